// FixBatchChebConv_38216619000238
// MI455X (gfx1250) — compile-verified
//
#include <hip/hip_runtime.h>
#include <hip/hip_bf16.h>

typedef __attribute__((ext_vector_type(2))) float v2f;
typedef __attribute__((ext_vector_type(8))) float v8f;

#define NB 50000
#define BB 8
#define CINB 64

// ---------------- graph preprocessing ----------------

__global__ void cheb_deg_kernel(const long long* __restrict__ ei, float* __restrict__ deg, int E) {
    int e = blockIdx.x * blockDim.x + threadIdx.x;
    if (e < E) {
        long long r = ei[e];
        long long c = ei[(size_t)E + e];
        if (r != c) atomicAdd(&deg[r], 1.0f);
    }
}

__global__ void cheb_dinv_kernel(float* __restrict__ deg, int n) {
    int i = blockIdx.x * blockDim.x + threadIdx.x;
    if (i < n) {
        float d = deg[i];
        deg[i] = (d > 0.0f) ? rsqrtf(d) : 0.0f;   // deg now holds dinv
    }
}

__global__ void cheb_lap_kernel(const long long* __restrict__ ei,
                                const float* __restrict__ dinv,
                                float* __restrict__ lap, int E) {
    int e = blockIdx.x * blockDim.x + threadIdx.x;
    if (e < E) {
        long long r = ei[e];
        long long c = ei[(size_t)E + e];
        lap[e] = (r != c) ? (-dinv[r] * dinv[c]) : 0.0f;
    }
}

// ---------------- sparse propagation: dst[row] += lap * src[col] ----------------
// one block (128 threads) per edge; each thread moves one float4 of the
// B*CIN = 512-float feature slab for that edge.
__global__ __launch_bounds__(128) void cheb_scatter_kernel(
        const float* __restrict__ src, float* __restrict__ dst,
        const float* __restrict__ lap, const long long* __restrict__ ei, int E) {
    int e = blockIdx.x;
    float l = lap[e];
    if (l == 0.0f) return;                  // self-loop / isolated: contributes nothing
    int r = (int)ei[e];
    int c = (int)ei[(size_t)E + e];
    int t = threadIdx.x;
    int b = t >> 4;                         // batch 0..7
    int q = t & 15;                         // float4 index within 64 channels
    const float4* s4 = (const float4*)(src + ((size_t)b * NB + c) * CINB);
    float4 v = s4[q];
    float* d = dst + ((size_t)b * NB + r) * CINB + q * 4;
    atomicAdd(d + 0, l * v.x);
    atomicAdd(d + 1, l * v.y);
    atomicAdd(d + 2, l * v.z);
    atomicAdd(d + 3, l * v.w);
}

// acc = 2*acc - x   (turns prop(tx1) accumulator into tx2)
__global__ void cheb_combine_kernel(float* __restrict__ acc, const float* __restrict__ x, int n4) {
    int i = blockIdx.x * blockDim.x + threadIdx.x;
    if (i < n4) {
        float4 a  = ((const float4*)acc)[i];
        float4 xv = ((const float4*)x)[i];
        a.x = 2.0f * a.x - xv.x;
        a.y = 2.0f * a.y - xv.y;
        a.z = 2.0f * a.z - xv.z;
        a.w = 2.0f * a.w - xv.w;
        ((float4*)acc)[i] = a;
    }
}

// ---------------- fused dense stage: out = T0@W0 + T1@W1 + T2@W2 + bias ----------------
// FP32 WMMA, 16x16x4. One wave owns a 16-row M-tile and all 64 output columns.
// A-fragment layout (ISA 7.12.2, 32-bit A 16x4): lane holds row lane%16,
// element j holds k = kb + 2*(lane/16) + j  -> contiguous float2 load.
// B-fragment (4x16, row striped across lanes): element j holds
// W[kb + 2*(lane/16) + j][nt*16 + lane%16].
// C/D (16x16 f32): element v -> row mbase + v + 8*(lane/16), col nt*16 + lane%16.

__device__ inline void cheb_accum_term(const float* __restrict__ T,
                                       const float* __restrict__ Wk,
                                       int mbase, int lm, int half, v8f acc[4]) {
    for (int kb = 0; kb < 64; kb += 4) {
        const float* ap = T + ((size_t)(mbase + lm)) * 64 + kb + 2 * half;
        v2f a = *(const v2f*)ap;
        int krow = kb + 2 * half;
#pragma unroll
        for (int nt = 0; nt < 4; ++nt) {
            v2f b;
            b.x = Wk[(krow + 0) * 64 + nt * 16 + lm];
            b.y = Wk[(krow + 1) * 64 + nt * 16 + lm];
            acc[nt] = __builtin_amdgcn_wmma_f32_16x16x4_f32(
                false, a, false, b, (short)0, acc[nt], false, false);
        }
    }
}

__global__ __launch_bounds__(256) void cheb_gemm_wmma_kernel(
        const float* __restrict__ T0, const float* __restrict__ T1,
        const float* __restrict__ T2, const float* __restrict__ W,
        const float* __restrict__ bias, float* __restrict__ out) {
    __shared__ float lds_w[3 * 64 * 64];   // 48 KB of weights in LDS
    for (int i = threadIdx.x; i < 3 * 64 * 64; i += 256) lds_w[i] = W[i];
    __syncthreads();

    const int lane = threadIdx.x & 31;
    const int wave = threadIdx.x >> 5;
    const int mbase = (blockIdx.x * 8 + wave) * 16;   // M = 400000 divisible by 128
    const int half = lane >> 4;
    const int lm   = lane & 15;

    v8f acc[4];
#pragma unroll
    for (int nt = 0; nt < 4; ++nt) acc[nt] = v8f{0, 0, 0, 0, 0, 0, 0, 0};

    cheb_accum_term(T0, &lds_w[0 * 4096], mbase, lm, half, acc);
    cheb_accum_term(T1, &lds_w[1 * 4096], mbase, lm, half, acc);
    cheb_accum_term(T2, &lds_w[2 * 4096], mbase, lm, half, acc);

#pragma unroll
    for (int nt = 0; nt < 4; ++nt) {
        float bv = bias[nt * 16 + lm];
#pragma unroll
        for (int v = 0; v < 8; ++v) {
            int row = mbase + v + 8 * half;
            out[(size_t)row * 64 + nt * 16 + lm] = acc[nt][v] + bv;
        }
    }
}

// ---------------- host-side orchestration ----------------

extern "C" void kernel_launch(void* const* d_in, const int* in_sizes, int n_in,
                              void* d_out, int out_size, void* d_ws, size_t ws_size,
                              hipStream_t stream) {
    (void)n_in; (void)out_size; (void)ws_size;

    const float*     x    = (const float*)d_in[0];          // [8, 50000, 64]
    const float*     W    = (const float*)d_in[1];          // [3, 64, 64]
    const float*     bias = (const float*)d_in[2];          // [64]
    const long long* ei   = (const long long*)d_in[3];      // [2, E] int64
    float*           out  = (float*)d_out;                  // [8, 50000, 64]

    const int E = in_sizes[3] / 2;
    const size_t feat = (size_t)BB * NB * CINB;             // 25.6M floats

    float* ws = (float*)d_ws;
    size_t lap_off = 65536;                                 // > N, 256-aligned
    size_t tx1_off = lap_off + (((size_t)E + 255) / 256) * 256;
    size_t acc_off = tx1_off + feat;                        // tx1 & acc contiguous

    float* deg = ws;
    float* lap = ws + lap_off;
    float* tx1 = ws + tx1_off;
    float* acc = ws + acc_off;

    // deterministic re-zero of atomic accumulation buffers every call
    hipMemsetAsync(deg, 0, (size_t)NB * sizeof(float), stream);
    hipMemsetAsync(tx1, 0, 2 * feat * sizeof(float), stream);   // covers tx1 + acc

    cheb_deg_kernel<<<(E + 255) / 256, 256, 0, stream>>>(ei, deg, E);
    cheb_dinv_kernel<<<(NB + 255) / 256, 256, 0, stream>>>(deg, NB);
    cheb_lap_kernel<<<(E + 255) / 256, 256, 0, stream>>>(ei, deg, lap, E);

    // tx1 = prop(x)
    cheb_scatter_kernel<<<E, 128, 0, stream>>>(x, tx1, lap, ei, E);
    // acc = prop(tx1); then acc = 2*acc - x  (== tx2)
    cheb_scatter_kernel<<<E, 128, 0, stream>>>(tx1, acc, lap, ei, E);
    cheb_combine_kernel<<<(int)(feat / 4 + 255) / 256, 256, 0, stream>>>(acc, x, (int)(feat / 4));

    // out = x@W0 + tx1@W1 + tx2@W2 + bias   (M = 400000 rows, 3125 blocks * 8 waves * 16 rows)
    cheb_gemm_wmma_kernel<<<(BB * NB) / 128, 256, 0, stream>>>(x, tx1, acc, W, bias, out);
}